// Descriptor_Net_11484742549547
// MI455X (gfx1250) — compile-verified
//
#include <hip/hip_runtime.h>
#include <math.h>

#define RAD_N 9
#define ELE_N 40
#define AZI_N 80
#define MVOX (RAD_N*ELE_N*AZI_N)   // 28800
#define NSAMP 30
#define BB 2
#define NP 1024
#define PI_F 3.14159265358979323846f
#define EPS_F 1e-5f

typedef __attribute__((ext_vector_type(2))) float v2f;
typedef __attribute__((ext_vector_type(8))) float v8f;

// ---------------------------------------------------------------------------
// Stage 1: neighbor gather + rotation. One wave per (b, m).
// Picks the first (lowest-index) 30 points with ||delta - vox||^2 <= 0.09,
// stores rotated rel vectors; masked slots = 0 (matches reference).
// ---------------------------------------------------------------------------
__global__ void __launch_bounds__(256) gather_kernel(
    const float* __restrict__ pts, float* __restrict__ rel)
{
  int wave = blockIdx.x * 8 + (threadIdx.x >> 5);
  int lane = threadIdx.x & 31;
  int b = wave / MVOX, m = wave % MVOX;
  int ri  = m / (ELE_N * AZI_N);
  int rem = m % (ELE_N * AZI_N);
  int ei = rem / AZI_N, ai = rem % AZI_N;

  float rv = ((float)ri + 0.5f) * (2.0f / (float)RAD_N);
  float el = ((float)ei + 0.5f) * (PI_F / (float)ELE_N);
  float az = ((float)ai + 0.5f) * (2.0f * PI_F / (float)AZI_N);
  float sel = sinf(el), cel = cosf(el);
  float vx = rv * sel * cosf(az);
  float vy = rv * sel * sinf(az);
  float vz = rv * cel;
  float v2 = vx * vx + vy * vy + vz * vz;

  float cx = pts[((size_t)b * NP + (NP - 1)) * 3 + 0];
  float cy = pts[((size_t)b * NP + (NP - 1)) * 3 + 1];
  float cz = pts[((size_t)b * NP + (NP - 1)) * 3 + 2];

  float ang = -(float)ai * (2.0f * PI_F / (float)AZI_N);
  float ca = cosf(ang), sa = sinf(ang);

  float* out = rel + (size_t)wave * NSAMP * 3;
  int count = 0;
  for (int chunk = 0; chunk < NP / 32; ++chunk) {
    if (count >= NSAMP) break;
    int n = chunk * 32 + lane;
    float dx = pts[((size_t)b * NP + n) * 3 + 0] - cx;
    float dy = pts[((size_t)b * NP + n) * 3 + 1] - cy;
    float dz = pts[((size_t)b * NP + n) * 3 + 2] - cz;
    // same algebraic form as the reference d2 expansion
    float d2 = (dx*dx + dy*dy + dz*dz) + v2 - 2.0f*(dx*vx + dy*vy + dz*vz);
    bool pred = d2 <= 0.09f;
    unsigned msk = (unsigned)__ballot(pred);          // wave32: low 32 bits
    int before = __popc(msk & ((1u << lane) - 1u));
    int slot = count + before;
    if (pred && slot < NSAMP) {
      float rx = dx - vx, ry = dy - vy, rz = dz - vz;
      out[slot * 3 + 0] = rx * ca + ry * sa;          // einsum with R[a]
      out[slot * 3 + 1] = ry * ca - rx * sa;
      out[slot * 3 + 2] = rz;
    }
    count += __popc(msk);
  }
  if (count > NSAMP) count = NSAMP;
  for (int s = count + lane; s < NSAMP; s += 32) {
    out[s * 3 + 0] = 0.f; out[s * 3 + 1] = 0.f; out[s * 3 + 2] = 0.f;
  }
}

// ---------------------------------------------------------------------------
// Stage 2a: per-block partial sums of y and y^2 for the raise layer BN stats.
// Deterministic (fixed grid, tree reduce, fixed-order final pass).
// ---------------------------------------------------------------------------
__global__ void __launch_bounds__(256) raise_stats_partial(
    const float* __restrict__ rel, const float* __restrict__ w,
    const float* __restrict__ br, float* __restrict__ part)
{
  __shared__ float wsm[48], bsm[16], red[256];
  int tid = threadIdx.x;
  if (tid < 48) wsm[tid] = w[tid];
  if (tid < 16) bsm[tid] = br[tid];
  __syncthreads();
  const size_t S = (size_t)BB * MVOX * NSAMP;
  float sum[16], sq[16];
  #pragma unroll
  for (int o = 0; o < 16; ++o) { sum[o] = 0.f; sq[o] = 0.f; }
  for (size_t i = (size_t)blockIdx.x * 256 + tid; i < S; i += (size_t)gridDim.x * 256) {
    float rx = rel[i * 3 + 0], ry = rel[i * 3 + 1], rz = rel[i * 3 + 2];
    #pragma unroll
    for (int o = 0; o < 16; ++o) {
      float yv = wsm[o*3]*rx + wsm[o*3+1]*ry + wsm[o*3+2]*rz + bsm[o];
      sum[o] += yv; sq[o] += yv * yv;
    }
  }
  for (int q = 0; q < 32; ++q) {
    red[tid] = (q < 16) ? sum[q] : sq[q - 16];
    __syncthreads();
    for (int off = 128; off > 0; off >>= 1) {
      if (tid < off) red[tid] += red[tid + off];
      __syncthreads();
    }
    if (tid == 0) part[(size_t)blockIdx.x * 32 + q] = red[0];
    __syncthreads();
  }
}

__global__ void raise_stats_final(const float* __restrict__ part, int nblk,
                                  float* __restrict__ stats)
{
  __shared__ float acc[32];
  int tid = threadIdx.x;
  if (tid < 32) {
    float s = 0.f;
    for (int b = 0; b < nblk; ++b) s += part[(size_t)b * 32 + tid];
    acc[tid] = s;
  }
  __syncthreads();
  if (tid < 16) {
    const float cnt = (float)((size_t)BB * MVOX * NSAMP);
    float mean = acc[tid] / cnt;
    stats[tid] = mean;                         // mean[16]
    stats[16 + tid] = acc[16 + tid] / cnt - mean * mean;  // var[16]
  }
}

// ---------------------------------------------------------------------------
// Stage 2b: raise layer apply: y = W*rel + b, BN, ReLU, max over 30 samples.
// Output x0: (B, 16, M) with M ordered (rad, ele, azi) == (D, H, W).
// ---------------------------------------------------------------------------
__global__ void __launch_bounds__(256) raise_apply_max(
    const float* __restrict__ rel, const float* __restrict__ w,
    const float* __restrict__ br, const float* __restrict__ g,
    const float* __restrict__ be, const float* __restrict__ stats,
    float* __restrict__ x0)
{
  __shared__ float wsm[48], bsm[16], scl[16], sft[16];
  int tid = threadIdx.x;
  if (tid < 48) wsm[tid] = w[tid];
  if (tid < 16) {
    bsm[tid] = br[tid];
    float mean = stats[tid], var = stats[16 + tid];
    float s = g[tid] * rsqrtf(var + EPS_F);
    scl[tid] = s;
    sft[tid] = be[tid] - mean * s;
  }
  __syncthreads();
  int i = blockIdx.x * 256 + tid;
  if (i >= BB * MVOX) return;
  float mx[16];
  #pragma unroll
  for (int o = 0; o < 16; ++o) mx[o] = -1e30f;
  const float* rp = rel + (size_t)i * NSAMP * 3;
  for (int s = 0; s < NSAMP; ++s) {
    float rx = rp[s*3], ry = rp[s*3+1], rz = rp[s*3+2];
    #pragma unroll
    for (int o = 0; o < 16; ++o) {
      float yv = wsm[o*3]*rx + wsm[o*3+1]*ry + wsm[o*3+2]*rz + bsm[o];
      float v = scl[o] * yv + sft[o];
      mx[o] = fmaxf(mx[o], v);
    }
  }
  int b = i / MVOX, m = i % MVOX;
  #pragma unroll
  for (int o = 0; o < 16; ++o)
    x0[((size_t)b * 16 + o) * MVOX + m] = fmaxf(mx[o], 0.f);  // relu(max)=max(relu)
}

// ---------------------------------------------------------------------------
// Stage 3: 3x3x3 conv (VALID depth, zero-pad ele, wrap azi) via f32 WMMA.
// One wave computes a tile of 16 azimuth positions x 32 output channels.
// K-order: (kd, ke, ka, ci in groups of 4) -> V_WMMA_F32_16X16X4_F32 steps.
// Weights staged per-kd slice in LDS (32*Cin*9 floats <= 36 KB).
// ---------------------------------------------------------------------------
__global__ void __launch_bounds__(256) conv3d_wmma(
    const float* __restrict__ in, const float* __restrict__ wgt,
    const float* __restrict__ bias, float* __restrict__ out,
    int Cin, int Din)
{
  __shared__ float wl[32 * 32 * 9];
  const int Dout = Din - 2;
  const int lane = threadIdx.x & 31;
  const int tile = blockIdx.x * 8 + (threadIdx.x >> 5);
  int at = tile % 5;      int t1 = tile / 5;
  int e  = t1 % ELE_N;    int t2 = t1 / ELE_N;
  int d  = t2 % Dout;     int b  = t2 / Dout;

  const int mrow = lane & 15;              // A: row M / B: col N / C: N
  const int koff = (lane >> 4) << 1;       // K sub-offset {0,1} vs {2,3}
  const int aout = at * 16 + mrow;         // output azimuth for this lane's A row

  float bias0 = bias[mrow];
  float bias1 = bias[16 + mrow];
  v8f c0, c1;
  #pragma unroll
  for (int j = 0; j < 8; ++j) { c0[j] = bias0; c1[j] = bias1; }

  const int HW = ELE_N * AZI_N;            // 3200
  const long cstr = (long)Din * HW;        // channel stride (floats)
  const float* inb = in + (long)b * Cin * cstr;
  const int wslice = 32 * Cin * 9;

  for (int kd = 0; kd < 3; ++kd) {
    __syncthreads();
    for (int idx = threadIdx.x; idx < wslice; idx += 256) {
      int kk = idx % 9; int t = idx / 9; int ci = t % Cin; int o = t / Cin;
      wl[idx] = wgt[(o * Cin + ci) * 27 + kd * 9 + kk];
    }
    __syncthreads();
    const float* ind = inb + (long)(d + kd) * HW;
    for (int ke = 0; ke < 3; ++ke) {
      int epos = e + ke - 1;
      bool eok = (epos >= 0) && (epos < ELE_N);
      int ec = eok ? epos : 0;
      for (int ka = 0; ka < 3; ++ka) {
        int apos = aout + ka - 1;
        if (apos < 0) apos += AZI_N;
        if (apos >= AZI_N) apos -= AZI_N;
        const float* ap = ind + ec * AZI_N + apos;
        const int kk = ke * 3 + ka;
        const float* wp0 = wl + (mrow * Cin + koff) * 9 + kk;
        const float* wp1 = wl + ((16 + mrow) * Cin + koff) * 9 + kk;
        for (int ci0 = 0; ci0 < Cin; ci0 += 4) {
          v2f A;
          if (eok) {
            A[0] = ap[(long)(ci0 + koff)     * cstr];
            A[1] = ap[(long)(ci0 + koff + 1) * cstr];
          } else { A[0] = 0.f; A[1] = 0.f; }
          v2f B0, B1;
          B0[0] = wp0[ci0 * 9];
          B0[1] = wp0[ci0 * 9 + 9];
          B1[0] = wp1[ci0 * 9];
          B1[1] = wp1[ci0 * 9 + 9];
          c0 = __builtin_amdgcn_wmma_f32_16x16x4_f32(false, A, false, B0,
                                                     (short)0, c0, false, false);
          c1 = __builtin_amdgcn_wmma_f32_16x16x4_f32(false, A, false, B1,
                                                     (short)0, c1, false, false);
        }
      }
    }
  }
  // D layout: VGPR j holds (M = j + 8*(lane>=16), N = lane&15) -> 8 consecutive
  // azimuth outputs per lane.
  const int Mb = (lane >> 4) * 8;
  long ob0 = ((((long)b * 32 + mrow)      * Dout + d) * ELE_N + e) * AZI_N + at * 16 + Mb;
  long ob1 = ((((long)b * 32 + 16 + mrow) * Dout + d) * ELE_N + e) * AZI_N + at * 16 + Mb;
  #pragma unroll
  for (int j = 0; j < 8; ++j) { out[ob0 + j] = c0[j]; out[ob1 + j] = c1[j]; }
}

// ---------------------------------------------------------------------------
// Per-channel BN stats for conv outputs (one block per channel, deterministic).
// ---------------------------------------------------------------------------
__global__ void __launch_bounds__(256) conv_stats(
    const float* __restrict__ y, int Dout, float* __restrict__ stats)
{
  __shared__ float rs[256], rq[256];
  int c = blockIdx.x, tid = threadIdx.x;
  size_t per = (size_t)Dout * ELE_N * AZI_N;
  float s = 0.f, q = 0.f;
  for (int b = 0; b < BB; ++b) {
    const float* p = y + ((size_t)b * 32 + c) * per;
    for (size_t i = tid; i < per; i += 256) { float v = p[i]; s += v; q += v * v; }
  }
  rs[tid] = s; rq[tid] = q; __syncthreads();
  for (int off = 128; off > 0; off >>= 1) {
    if (tid < off) { rs[tid] += rs[tid + off]; rq[tid] += rq[tid + off]; }
    __syncthreads();
  }
  if (tid == 0) {
    float cnt = (float)(BB * per);
    float mean = rs[0] / cnt;
    stats[c] = mean;
    stats[32 + c] = rq[0] / cnt - mean * mean;
  }
}

__global__ void __launch_bounds__(256) bnrelu(
    float* __restrict__ y, const float* __restrict__ g,
    const float* __restrict__ be, const float* __restrict__ stats,
    int Dout, size_t tot)
{
  size_t i = (size_t)blockIdx.x * 256 + threadIdx.x;
  if (i >= tot) return;
  size_t per = (size_t)Dout * ELE_N * AZI_N;
  int c = (int)((i / per) % 32);
  float v = g[c] * (y[i] - stats[c]) * rsqrtf(stats[32 + c] + EPS_F) + be[c];
  y[i] = fmaxf(v, 0.f);
}

// ---------------------------------------------------------------------------
// Final: max over (ele, azi) for each (b, channel). Output 64 floats.
// ---------------------------------------------------------------------------
__global__ void __launch_bounds__(256) final_max(
    const float* __restrict__ x, float* __restrict__ outp)
{
  __shared__ float red[256];
  int bo = blockIdx.x, tid = threadIdx.x;
  const float* p = x + (size_t)bo * (ELE_N * AZI_N);
  float mx = -1e30f;
  for (int i = tid; i < ELE_N * AZI_N; i += 256) mx = fmaxf(mx, p[i]);
  red[tid] = mx; __syncthreads();
  for (int off = 128; off > 0; off >>= 1) {
    if (tid < off) red[tid] = fmaxf(red[tid], red[tid + off]);
    __syncthreads();
  }
  if (tid == 0) outp[bo] = red[0];
}

// ---------------------------------------------------------------------------
extern "C" void kernel_launch(void* const* d_in, const int* in_sizes, int n_in,
                              void* d_out, int out_size, void* d_ws, size_t ws_size,
                              hipStream_t stream) {
  (void)in_sizes; (void)n_in; (void)out_size; (void)ws_size;
  const float* pts     = (const float*)d_in[0];
  const float* w_raise = (const float*)d_in[1];
  const float* b_raise = (const float*)d_in[2];
  const float* g0      = (const float*)d_in[3];
  const float* be0     = (const float*)d_in[4];
  const float *cw[4], *cb[4], *cg[4], *cbe[4];
  for (int i = 0; i < 4; ++i) {
    cw[i]  = (const float*)d_in[5 + i * 4 + 0];
    cb[i]  = (const float*)d_in[5 + i * 4 + 1];
    cg[i]  = (const float*)d_in[5 + i * 4 + 2];
    cbe[i] = (const float*)d_in[5 + i * 4 + 3];
  }

  // Workspace layout (floats). Total ~9.40M floats (~38 MB).
  float* ws    = (float*)d_ws;
  float* rel   = ws;                                          // B*M*30*3
  float* x0    = rel  + (size_t)BB * MVOX * NSAMP * 3;        // B*16*M
  float* y1    = x0   + (size_t)BB * 16 * MVOX;               // B*32*7*3200
  float* y2    = y1   + (size_t)BB * 32 * 7 * 3200;           // B*32*5*3200
  float* y3    = y2   + (size_t)BB * 32 * 5 * 3200;           // B*32*3*3200
  float* y4    = y3   + (size_t)BB * 32 * 3 * 3200;           // B*32*1*3200
  float* part  = y4   + (size_t)BB * 32 * 1 * 3200;           // 512*32
  float* stats = part + (size_t)512 * 32;                     // 64

  // Stage 1: gather + rotate
  gather_kernel<<<(BB * MVOX) / 8, 256, 0, stream>>>(pts, rel);

  // Stage 2: raise MLP stats + apply + max
  raise_stats_partial<<<512, 256, 0, stream>>>(rel, w_raise, b_raise, part);
  raise_stats_final<<<1, 64, 0, stream>>>(part, 512, stats);
  raise_apply_max<<<(BB * MVOX + 255) / 256, 256, 0, stream>>>(
      rel, w_raise, b_raise, g0, be0, stats, x0);

  // Stage 3: four conv+BN+ReLU layers (WMMA implicit GEMM)
  const float* cin = x0;
  int Cin = 16, Din = RAD_N;
  float* ybuf[4] = {y1, y2, y3, y4};
  for (int l = 0; l < 4; ++l) {
    int Dout = Din - 2;
    int tiles = BB * Dout * ELE_N * 5;    // 16-wide azimuth tiles
    conv3d_wmma<<<tiles / 8, 256, 0, stream>>>(cin, cw[l], cb[l], ybuf[l], Cin, Din);
    conv_stats<<<32, 256, 0, stream>>>(ybuf[l], Dout, stats);
    size_t tot = (size_t)BB * 32 * Dout * ELE_N * AZI_N;
    bnrelu<<<(int)((tot + 255) / 256), 256, 0, stream>>>(
        ybuf[l], cg[l], cbe[l], stats, Dout, tot);
    cin = ybuf[l]; Cin = 32; Din = Dout;
  }

  // Stage 4: global max over (ele, azi) -> (B, 32, 1, 1)
  final_max<<<BB * 32, 256, 0, stream>>>(y4, (float*)d_out);
}